// AttributeClassifierHeaders_83580063580200
// MI455X (gfx1250) — compile-verified
//
#include <hip/hip_runtime.h>
#include <math.h>

typedef float v2f __attribute__((ext_vector_type(2)));
typedef float v8f __attribute__((ext_vector_type(8)));
typedef __attribute__((__vector_size__(4 * sizeof(int)))) int v4i;
typedef __attribute__((address_space(1))) v4i gv4i;   // global int4
typedef __attribute__((address_space(3))) v4i lv4i;   // LDS int4

#define IN_DIM 2048
#define HID    1024
#define ATT    40
#define BATCH  4096

#define BM 256            // rows per block
#define BN 128            // hidden cols per block
#define KC 64             // K chunk staged in LDS
#define NCHUNKS (IN_DIM / KC)   // 32
#define NCH     (HID / BN)      // 8 partial chunks per (b,a)
#define THREADS 256
#define WAVES   8

#define XPITCH 68         // KC+4: 16B-aligned rows, conflict-free ds_b64 A-frags
#define WPITCH 136        // BN+8: 16B-aligned rows, conflict-free ds_b32 B-frags
#define XBUF   (BM * XPITCH)    // 17408 floats
#define WBUF   (KC * WPITCH)    // 8704 floats
#define BUFSZ  (XBUF + WBUF)    // per double-buffer half

// ---------------- CDNA5 async global->LDS staging (with sync fallback) ------
#if __has_builtin(__builtin_amdgcn_global_load_async_to_lds_b128)
__device__ __forceinline__ void copy16_g2l(const float* g, float* l) {
    __builtin_amdgcn_global_load_async_to_lds_b128(
        (gv4i*)(unsigned long long)(const void*)g,
        (lv4i*)(unsigned)(unsigned long long)(void*)l,
        0, 0);
}
#else
__device__ __forceinline__ void copy16_g2l(const float* g, float* l) {
    *reinterpret_cast<float4*>(l) = *reinterpret_cast<const float4*>(g);
}
#endif

template <int N>
__device__ __forceinline__ void wait_async() {
#if __has_builtin(__builtin_amdgcn_s_wait_asynccnt)
    __builtin_amdgcn_s_wait_asynccnt(N);
#else
    asm volatile("s_wait_asynccnt %0" :: "n"(N) : "memory");
#endif
}

__device__ __forceinline__ v8f wmma4(v2f a, v2f b, v8f c) {
    return __builtin_amdgcn_wmma_f32_16x16x4_f32(false, a, false, b, (short)0, c,
                                                 false, false);
}

// stage one (x 256xKC, W1 KCxBN) chunk into an LDS buffer; 6 async ops / wave
__device__ __forceinline__ void stage_chunk(const float* __restrict__ x_blk,
                                            const float* __restrict__ w_blk,
                                            int kc, float* xbuf, float* wbuf,
                                            int tid)
{
    #pragma unroll
    for (int it = 0; it < 4; ++it) {               // 1024 float4 tasks for x
        int t = tid + it * THREADS;
        int row = t >> 2, j = t & 3;
        copy16_g2l(x_blk + (size_t)row * IN_DIM + kc + 4 * j,
                   xbuf + row * XPITCH + 4 * j);
    }
    #pragma unroll
    for (int it = 0; it < 2; ++it) {               // 512 float4 tasks for W1
        int t = tid + it * THREADS;
        int kr = t >> 3, j = t & 7;
        copy16_g2l(w_blk + (size_t)(kc + kr) * HID + 4 * j,
                   wbuf + kr * WPITCH + 4 * j);
    }
}

// Block: (head a, 256 rows, 128 hidden cols). Wave w owns row-subtiles {2w,2w+1}
// across all 8 n-tiles -> complete partial dot for its 32 rows, written to ws.
__global__ __launch_bounds__(THREADS)
void attn_heads_gemm_kernel(const float* __restrict__ x,
                            const float* __restrict__ W1,
                            const float* __restrict__ b1,
                            const float* __restrict__ W2,
                            float* __restrict__ ws)
{
    __shared__ float smem[2 * BUFSZ];              // ~204 KB of the 320 KB WGP LDS

    const int m0   = blockIdx.x * BM;
    const int ncid = blockIdx.y;
    const int n0   = ncid * BN;
    const int a    = blockIdx.z;
    const int tid  = threadIdx.x;
    const int wave = tid >> 5;
    const int lane = tid & 31;
    const int l16  = lane & 15;
    const int kh   = lane >> 4;                    // K-half within 16x16x4 step

    const float* x_blk = x + (size_t)m0 * IN_DIM;
    const float* w_blk = W1 + (size_t)a * IN_DIM * HID + n0;

    v8f acc[16];                                   // [ms*8 + nt]
    #pragma unroll
    for (int i = 0; i < 16; ++i) acc[i] = v8f{0.f,0.f,0.f,0.f,0.f,0.f,0.f,0.f};

    stage_chunk(x_blk, w_blk, 0, smem, smem + XBUF, tid);

    for (int c = 0; c < NCHUNKS; ++c) {
        float* xb = smem + (c & 1) * BUFSZ;
        float* wb = xb + XBUF;
        if (c + 1 < NCHUNKS) {
            float* xn = smem + ((c + 1) & 1) * BUFSZ;
            stage_chunk(x_blk, w_blk, (c + 1) * KC, xn, xn + XBUF, tid);
            wait_async<6>();                       // chunk c landed (ours)
        } else {
            wait_async<0>();
        }
        __syncthreads();                           // everyone's chunk c landed

        const float* A0 = xb + ((2 * wave)     * 16 + l16) * XPITCH + 2 * kh;
        const float* A1 = xb + ((2 * wave + 1) * 16 + l16) * XPITCH + 2 * kh;
        const float* B0 = wb + 2 * kh * WPITCH + l16;

        #pragma unroll 4
        for (int k = 0; k < KC; k += 4) {
            v2f af0 = *reinterpret_cast<const v2f*>(A0 + k);   // ds_load_b64
            v2f af1 = *reinterpret_cast<const v2f*>(A1 + k);
            const float* Bk = B0 + k * WPITCH;
            #pragma unroll
            for (int nt = 0; nt < 8; ++nt) {
                v2f bf;
                bf.x = Bk[nt * 16];
                bf.y = Bk[nt * 16 + WPITCH];
                acc[nt]     = wmma4(af0, bf, acc[nt]);
                acc[8 + nt] = wmma4(af1, bf, acc[8 + nt]);
            }
        }
        __syncthreads();                           // done reading xb/wb
    }

    // ---- fused epilogue: bias + exact gelu + dot(W2) over this N-chunk ----
    float psum[2][8];
    #pragma unroll
    for (int ms = 0; ms < 2; ++ms)
        #pragma unroll
        for (int v = 0; v < 8; ++v) psum[ms][v] = 0.f;

    #pragma unroll
    for (int nt = 0; nt < 8; ++nt) {
        const int n  = n0 + nt * 16 + l16;
        const float bias = b1[a * HID + n];
        const float w2v  = W2[a * HID + n];
        #pragma unroll
        for (int ms = 0; ms < 2; ++ms) {
            #pragma unroll
            for (int v = 0; v < 8; ++v) {
                float h = acc[ms * 8 + nt][v] + bias;
                h = 0.5f * h * (1.0f + erff(h * 0.70710678118654752f));
                psum[ms][v] += h * w2v;
            }
        }
    }
    // reduce the 16 column-lanes of each half-wave (masks 1..8 never cross halves)
    #pragma unroll
    for (int ms = 0; ms < 2; ++ms)
        #pragma unroll
        for (int v = 0; v < 8; ++v) {
            float s = psum[ms][v];
            #pragma unroll
            for (int m = 1; m < 16; m <<= 1) s += __shfl_xor(s, m, 32);
            psum[ms][v] = s;
        }

    if (l16 == 0) {                                // lanes 0 & 16 write 32 rows/wave
        #pragma unroll
        for (int ms = 0; ms < 2; ++ms)
            #pragma unroll
            for (int v = 0; v < 8; ++v) {
                int row = m0 + (2 * wave + ms) * 16 + v + 8 * kh;
                ws[((size_t)row * ATT + a) * NCH + ncid] = psum[ms][v];
            }
    }
}

__global__ __launch_bounds__(THREADS)
void attn_heads_finalize_kernel(const float* __restrict__ ws,
                                const float* __restrict__ b2,
                                float* __restrict__ out)
{
    int i = blockIdx.x * blockDim.x + threadIdx.x;   // i = b*ATT + a
    if (i < BATCH * ATT) {
        int a = i % ATT;
        float s = 0.f;
        #pragma unroll
        for (int c = 0; c < NCH; ++c) s += ws[(size_t)i * NCH + c];
        s += b2[a];
        out[i] = 1.0f / (1.0f + expf(-s));
    }
}

extern "C" void kernel_launch(void* const* d_in, const int* in_sizes, int n_in,
                              void* d_out, int out_size, void* d_ws, size_t ws_size,
                              hipStream_t stream) {
    const float* x  = (const float*)d_in[0];
    const float* W1 = (const float*)d_in[1];
    const float* b1 = (const float*)d_in[2];
    const float* W2 = (const float*)d_in[3];
    const float* b2 = (const float*)d_in[4];
    float* out = (float*)d_out;
    float* ws  = (float*)d_ws;     // BATCH*ATT*NCH floats = 5.24 MB partials

    dim3 grid1(BATCH / BM, NCH, ATT);
    attn_heads_gemm_kernel<<<grid1, dim3(THREADS), 0, stream>>>(x, W1, b1, W2, ws);

    dim3 grid2((BATCH * ATT + THREADS - 1) / THREADS);
    attn_heads_finalize_kernel<<<grid2, dim3(THREADS), 0, stream>>>(ws, b2, out);
}